// MaxAggregator_65644280152900
// MI455X (gfx1250) — compile-verified
//
#include <hip/hip_runtime.h>
#include <hip/hip_bf16.h>
#include <math.h>

// MaxAggregator: out[n, f] = max over (s, f') of features[neigh_idx[n, s], f']
//   features : [V=100000, F=128] fp32
//   neigh_idx: [N=100000, S=16]  int
//   out      : [N, F] fp32 (scalar row-max broadcast across the row)
//
// MI455X strategy: one wave32 per node. Stage the 16 gathered 512-byte rows
// into LDS with GLOBAL_LOAD_ASYNC_TO_LDS_B128 (ASYNCcnt path, 16 copies in
// flight per wave). Async loads complete in order, so we split the wait:
// asynccnt<=8 -> reduce rows 0..7 while rows 8..15 are in flight, then
// asynccnt<=0 -> reduce rows 8..15. Row indices are broadcast via v_readlane
// (SGPR) instead of ds_bpermute. Finish with a wave32 shfl_xor max tree and a
// coalesced float4 broadcast store.
//
// Roofline: ~6.4 MB idx + ~819 MB gathered rows (feature table is 51.2 MB ->
// L2-resident in the 192 MB L2) + 51 MB stores; pure bandwidth problem, WMMA
// inapplicable (max-reduce). The CDNA5-specific win is the ASYNCcnt copy path.

#define NODES_PER_BLOCK 8   // 8 waves * 32 lanes = 256 threads
#define S_NEIGH 16
#define FDIM 128

typedef int v4i __attribute__((ext_vector_type(4)));
typedef __attribute__((address_space(1))) v4i* gptr_v4i;
typedef __attribute__((address_space(3))) v4i* lptr_v4i;

__device__ __forceinline__ void async_copy_b128(const void* gsrc, void* lds_dst) {
#if __has_builtin(__builtin_amdgcn_global_load_async_to_lds_b128)
  __builtin_amdgcn_global_load_async_to_lds_b128(
      (gptr_v4i)(void*)gsrc, (lptr_v4i)lds_dst, 0, 0);
#else
  asm volatile("global_load_async_to_lds_b128 %0, %1, off"
               :: "v"((lptr_v4i)lds_dst), "v"((gptr_v4i)(void*)gsrc)
               : "memory");
#endif
}

template <int CNT>
__device__ __forceinline__ void wait_async() {
#if __has_builtin(__builtin_amdgcn_s_wait_asynccnt)
  __builtin_amdgcn_s_wait_asynccnt(CNT);
#else
  asm volatile("s_wait_asynccnt %0" :: "i"(CNT) : "memory");
#endif
  asm volatile("" ::: "memory");   // keep LDS reads ordered after the wait
}

__device__ __forceinline__ float max4(const float4 v) {
  return fmaxf(fmaxf(v.x, v.y), fmaxf(v.z, v.w));
}

__global__ __launch_bounds__(NODES_PER_BLOCK * 32)
void maxagg_kernel(const float* __restrict__ features,
                   const int*   __restrict__ neigh_idx,
                   float* __restrict__ out,
                   int N) {
  // 8 waves * 16 rows * 128 floats * 4B = 64 KB LDS
  __shared__ float lds[NODES_PER_BLOCK * S_NEIGH * FDIM];

  const int lane = threadIdx.x & 31;
  const int wave = threadIdx.x >> 5;
  const int node = blockIdx.x * NODES_PER_BLOCK + wave;
  if (node >= N) return;          // wave-uniform: EXEC stays all-ones per wave

  float* slot = &lds[wave * (S_NEIGH * FDIM)];

  // Lanes 0..15 each hold one neighbor index for this node.
  int myIdx = 0;
  if (lane < S_NEIGH) myIdx = neigh_idx[node * S_NEIGH + lane];

  // Issue 16 async row copies: row k is 512B = 32 lanes x 16B.
  // Lane l copies bytes [l*16, l*16+16) of row k -> conflict-free LDS layout.
  // Row index broadcast via readlane -> SGPR row base, SADDR-form async loads.
#pragma unroll
  for (int k = 0; k < S_NEIGH; ++k) {
    const int row = __builtin_amdgcn_readlane(myIdx, k);
    const float* gsrc = features + (size_t)row * FDIM + lane * 4;
    float* ldst = slot + k * FDIM + lane * 4;
    async_copy_b128(gsrc, ldst);
  }

  float m = -INFINITY;

  // Rows complete in issue order: front half ready once asynccnt <= 8.
  wait_async<8>();
#pragma unroll
  for (int k = 0; k < S_NEIGH / 2; ++k)
    m = fmaxf(m, max4(*(const float4*)(slot + k * FDIM + lane * 4)));

  wait_async<0>();
#pragma unroll
  for (int k = S_NEIGH / 2; k < S_NEIGH; ++k)
    m = fmaxf(m, max4(*(const float4*)(slot + k * FDIM + lane * 4)));

  // wave32 max tree
#pragma unroll
  for (int off = 16; off >= 1; off >>= 1)
    m = fmaxf(m, __shfl_xor(m, off, 32));

  // Broadcast scalar into the 128-wide output row: lane l writes floats [4l, 4l+4).
  float4 o = make_float4(m, m, m, m);
  *(float4*)(out + (size_t)node * FDIM + lane * 4) = o;
}

extern "C" void kernel_launch(void* const* d_in, const int* in_sizes, int n_in,
                              void* d_out, int out_size, void* d_ws, size_t ws_size,
                              hipStream_t stream) {
  const float* features  = (const float*)d_in[0];
  const int*   neigh_idx = (const int*)d_in[1];
  float*       out       = (float*)d_out;

  const int N = in_sizes[1] / S_NEIGH;   // neigh_idx is [N, 16]
  const int grid = (N + NODES_PER_BLOCK - 1) / NODES_PER_BLOCK;

  maxagg_kernel<<<grid, NODES_PER_BLOCK * 32, 0, stream>>>(features, neigh_idx, out, N);
}